// DuelingCNNSurakartaNet_43576738186082
// MI455X (gfx1250) — compile-verified
//
#include <hip/hip_runtime.h>
#include <hip/hip_bf16.h>

// CDNA5 / gfx1250, wave32. All GEMMs via V_WMMA_F32_16X16X4_F32 (exact fp32).

typedef __attribute__((ext_vector_type(2))) float v2f;
typedef __attribute__((ext_vector_type(8))) float v8f;

#define WPB 8            // waves per 256-thread block

// conv: K-chunk must be a multiple of 36 (= 9 taps * 4 wmma-K); LDS row stride
// padded so 34*row mod 64 covers all 32 rows -> conflict-free ds_load_b64.
#define CONV_KC   288
#define AS_STRIDE 290    // 32*290*4 = 37120 B

// fc: K is always a multiple of 256 here (4608, 256). W-tile stride padded so
// 2*n mod 64 distinct -> conflict-free ds_load_b64.
#define FC_KC     256
#define WS_STRIDE 258    // 32*258*4 = 33024 B

__device__ __forceinline__ v8f wmma_f32_k4(v2f a, v2f b, v8f c) {
  // D = A(16x4) * B(4x16) + C(16x16), fp32 throughout.
  return __builtin_amdgcn_wmma_f32_16x16x4_f32(false, a, false, b, (short)0, c,
                                               false, false);
}

// ---------------------------------------------------------------------------
// Implicit-GEMM conv3x3 SAME on 6x6 boards, fused conv-bias + BN(eval) + ReLU.
// M = B*36 rows (batch,pos), N = Cout, K = Cin*9.  x: (B,Cin,36) y: (B,Cout,36)
// Block = 32 output rows; im2col tile staged in LDS once, shared by 8 waves
// (one 16-channel N-tile each, two 16-row M-subtiles each).
// ---------------------------------------------------------------------------
__global__ __launch_bounds__(256)
void conv_bn_relu_wmma(const float* __restrict__ x, const float* __restrict__ w,
                       const float* __restrict__ cb,
                       const float* __restrict__ bng, const float* __restrict__ bnb,
                       const float* __restrict__ bnm, const float* __restrict__ bnv,
                       float* __restrict__ y, int Cin, int Cout)
{
  __shared__ float As[32 * AS_STRIDE];
  const int K     = Cin * 9;
  const int lane  = threadIdx.x & 31;
  const int wave  = threadIdx.x >> 5;      // == nTile
  const int mBase = blockIdx.x * 32;

  // --- staging coordinates (row fixed per thread: 256 % 32 == 0) ---
  const int  srow = threadIdx.x & 31;
  const int  sm   = mBase + srow;
  const int  sb   = sm / 36;
  const int  sp   = sm - sb * 36;
  const int  spy  = sp / 6;
  const int  spx  = sp - spy * 6;
  const float* xrow = x + (size_t)sb * Cin * 36;

  // --- compute coordinates ---
  const int  nSub   = lane & 15;
  const int  kHalf  = (lane >> 4) << 1;    // 0 or 2
  const bool active = (wave * 16) < Cout;  // conv1: waves 4..7 only stage
  const int  gn     = wave * 16 + nSub;    // output channel
  const float* wrow = w + (size_t)gn * K;

  const v8f zero = {0.f, 0.f, 0.f, 0.f, 0.f, 0.f, 0.f, 0.f};
  v8f acc[2] = {zero, zero};

  for (int kc = 0; kc < K; kc += CONV_KC) {
    const int kchunk = (K - kc < CONV_KC) ? (K - kc) : CONV_KC;
    const int cChunk = kchunk / 9;
    const int cBase  = kc / 9;
    __syncthreads();
    // stage im2col: thread handles (srow, cr) pairs, 9 taps each, no divisions
    for (int cr = wave; cr < cChunk; cr += WPB) {
      const float* xc  = xrow + (size_t)(cBase + cr) * 36;
      float*       dst = As + srow * AS_STRIDE + cr * 9;
      #pragma unroll
      for (int dy = 0; dy < 3; ++dy) {
        const int iy = spy + dy - 1;
        #pragma unroll
        for (int dx = 0; dx < 3; ++dx) {
          const int ix = spx + dx - 1;
          float v = 0.f;
          if ((unsigned)iy < 6u && (unsigned)ix < 6u) v = xc[iy * 6 + ix];
          dst[dy * 3 + dx] = v;
        }
      }
    }
    __syncthreads();
    if (active) {
      const float* wk = wrow + kc;
      for (int k0 = 0; k0 < kchunk; k0 += 4) {
        const int ka = k0 + kHalf;
        const v2f bf = *(const v2f*)(wk + ka);                         // global
        const v2f a0 = *(const v2f*)(&As[nSub * AS_STRIDE + ka]);      // LDS
        const v2f a1 = *(const v2f*)(&As[(16 + nSub) * AS_STRIDE + ka]);
        acc[0] = wmma_f32_k4(a0, bf, acc[0]);
        acc[1] = wmma_f32_k4(a1, bf, acc[1]);
      }
    }
  }

  if (active) {
    const float scale = bng[gn] * rsqrtf(bnv[gn] + 1e-5f);
    const float shift = (cb[gn] - bnm[gn]) * scale + bnb[gn];
    const int   mAdd  = (lane >> 4) << 3;
    #pragma unroll
    for (int t = 0; t < 2; ++t) {
      const int m0 = mBase + t * 16 + mAdd;
      int bb = m0 / 36;
      int pp = m0 - bb * 36;
      #pragma unroll
      for (int r = 0; r < 8; ++r) {
        float v = acc[t][r] * scale + shift;
        y[((size_t)bb * Cout + gn) * 36 + pp] = v > 0.f ? v : 0.f;
        if (++pp == 36) { pp = 0; ++bb; }
      }
    }
  }
}

// ---------------------------------------------------------------------------
// Generic FC:  Y[M,N] = A[M,K] @ W[N,K]^T + bias  (+ optional one-hot column
// gather: extraW[n*ldw + sel[m]] — folds the argmax concat into the epilogue).
// Requires K % 256 == 0 (true here: 4608, 256). Each wave: 16x32 output tile.
// ---------------------------------------------------------------------------
__global__ __launch_bounds__(256)
void fc_wmma(const float* __restrict__ A, const float* __restrict__ W,
             const float* __restrict__ bias,
             const float* __restrict__ extraW, const int* __restrict__ sel,
             float* __restrict__ Y, int N, int K, int ldw)
{
  __shared__ float Ws[32 * WS_STRIDE];
  const int nTile = blockIdx.y;            // 32 columns per tile
  const int lane  = threadIdx.x & 31;
  const int wave  = threadIdx.x >> 5;
  const int mBase = (blockIdx.x * WPB + wave) * 16;
  const int kHalf = (lane >> 4) << 1;
  const int nSub  = lane & 15;
  const float* arow = A + (size_t)(mBase + nSub) * K;

  const v8f zero = {0.f, 0.f, 0.f, 0.f, 0.f, 0.f, 0.f, 0.f};
  v8f acc[2] = {zero, zero};

  for (int kc = 0; kc < K; kc += FC_KC) {
    __syncthreads();
    #pragma unroll 8
    for (int n = 0; n < 32; ++n) {
      const int gnn = nTile * 32 + n;
      Ws[n * WS_STRIDE + threadIdx.x] =
          (gnn < N) ? W[(size_t)gnn * ldw + kc + threadIdx.x] : 0.f;
    }
    __syncthreads();
    const float* ap = arow + kc;
    for (int k0 = 0; k0 < FC_KC; k0 += 4) {
      const int ka = k0 + kHalf;
      const v2f a  = *(const v2f*)(ap + ka);                            // global
      const v2f b0 = *(const v2f*)(&Ws[nSub * WS_STRIDE + ka]);         // LDS
      const v2f b1 = *(const v2f*)(&Ws[(16 + nSub) * WS_STRIDE + ka]);
      acc[0] = wmma_f32_k4(a, b0, acc[0]);
      acc[1] = wmma_f32_k4(a, b1, acc[1]);
    }
  }

  const int mAdd = (lane >> 4) << 3;
  #pragma unroll
  for (int t = 0; t < 2; ++t) {
    const int gn = nTile * 32 + t * 16 + nSub;
    if (gn < N) {
      const float add = bias[gn];
      #pragma unroll
      for (int r = 0; r < 8; ++r) {
        const int m = mBase + mAdd + r;
        float v = acc[t][r] + add;
        if (sel) v += extraW[(size_t)gn * ldw + sel[m]];
        Y[(size_t)m * N + gn] = v;
      }
    }
  }
}

// ---------------------------------------------------------------------------
// In-place LayerNorm(256) + ReLU, one wave per row (wave32 shuffle reduce).
// ---------------------------------------------------------------------------
__global__ __launch_bounds__(256)
void ln_relu(float* __restrict__ Y, const float* __restrict__ g,
             const float* __restrict__ b)
{
  const int row  = blockIdx.x * WPB + (threadIdx.x >> 5);
  const int lane = threadIdx.x & 31;
  float* y = Y + (size_t)row * 256;
  float v[8];
  float s = 0.f;
  #pragma unroll
  for (int i = 0; i < 8; ++i) { v[i] = y[lane + 32 * i]; s += v[i]; }
  #pragma unroll
  for (int off = 16; off; off >>= 1) s += __shfl_xor(s, off, 32);
  const float mu = s * (1.f / 256.f);
  float vs = 0.f;
  #pragma unroll
  for (int i = 0; i < 8; ++i) { const float d = v[i] - mu; vs += d * d; }
  #pragma unroll
  for (int off = 16; off; off >>= 1) vs += __shfl_xor(vs, off, 32);
  const float inv = rsqrtf(vs * (1.f / 256.f) + 1e-5f);
  #pragma unroll
  for (int i = 0; i < 8; ++i) {
    const int c = lane + 32 * i;
    const float o = (v[i] - mu) * inv * g[c] + b[c];
    y[c] = o > 0.f ? o : 0.f;
  }
}

// ---------------------------------------------------------------------------
// argmax over 36 logits per row (first-max tie-break, like jnp.argmax).
// ---------------------------------------------------------------------------
__global__ __launch_bounds__(256)
void argmax36(const float* __restrict__ L, int* __restrict__ sel)
{
  const int row  = blockIdx.x * WPB + (threadIdx.x >> 5);
  const int lane = threadIdx.x & 31;
  const float* l = L + (size_t)row * 36;
  float v  = l[lane];
  int  idx = lane;
  if (lane < 4) {
    const float v2 = l[32 + lane];
    if (v2 > v) { v = v2; idx = 32 + lane; }   // strict: keep earlier on tie
  }
  #pragma unroll
  for (int off = 16; off; off >>= 1) {
    const float ov = __shfl_xor(v, off, 32);
    const int   oi = __shfl_xor(idx, off, 32);
    if (ov > v || (ov == v && oi < idx)) { v = ov; idx = oi; }
  }
  if (lane == 0) sel[row] = idx;
}

// ---------------------------------------------------------------------------
// vfc2: val[m] = dot(H[m,:256], w) + b   (one wave per row)
// ---------------------------------------------------------------------------
__global__ __launch_bounds__(256)
void vfc2_dot(const float* __restrict__ H, const float* __restrict__ w,
              const float* __restrict__ b, float* __restrict__ val)
{
  const int row  = blockIdx.x * WPB + (threadIdx.x >> 5);
  const int lane = threadIdx.x & 31;
  const float* h = H + (size_t)row * 256;
  float s = 0.f;
  #pragma unroll
  for (int i = 0; i < 8; ++i) s += h[lane + 32 * i] * w[lane + 32 * i];
  #pragma unroll
  for (int off = 16; off; off >>= 1) s += __shfl_xor(s, off, 32);
  if (lane == 0) val[row] = s + b[0];
}

// ---------------------------------------------------------------------------
// Dueling combine: q[b, i*36+j] = val[b] + (p[i]-mean(p)) + (m[j]-mean(m))
// one wave per batch row; advantages staged in LDS.
// ---------------------------------------------------------------------------
__global__ __launch_bounds__(256)
void dueling_combine(const float* __restrict__ plog, const float* __restrict__ mlog,
                     const float* __restrict__ val, float* __restrict__ out)
{
  __shared__ float pb[WPB][36];
  __shared__ float mb[WPB][36];
  const int wave = threadIdx.x >> 5;
  const int lane = threadIdx.x & 31;
  const int row  = blockIdx.x * WPB + wave;
  const float* p = plog + (size_t)row * 36;
  const float* m = mlog + (size_t)row * 36;
  const float pa = p[lane], ma = m[lane];
  const float pe = (lane < 4) ? p[32 + lane] : 0.f;
  const float me = (lane < 4) ? m[32 + lane] : 0.f;
  float ps = pa + pe, ms = ma + me;
  #pragma unroll
  for (int off = 16; off; off >>= 1) {
    ps += __shfl_xor(ps, off, 32);
    ms += __shfl_xor(ms, off, 32);
  }
  const float pmean = ps * (1.f / 36.f);
  const float mmean = ms * (1.f / 36.f);
  pb[wave][lane] = pa - pmean;
  mb[wave][lane] = ma - mmean;
  if (lane < 4) { pb[wave][32 + lane] = pe - pmean; mb[wave][32 + lane] = me - mmean; }
  __syncthreads();
  const float v = val[row];
  float* o = out + (size_t)row * 1296;
  for (int t = lane; t < 1296; t += 32) {
    const int i = t / 36;
    const int j = t - i * 36;
    o[t] = v + pb[wave][i] + mb[wave][j];
  }
}

// ---------------------------------------------------------------------------
extern "C" void kernel_launch(void* const* d_in, const int* in_sizes, int n_in,
                              void* d_out, int out_size, void* d_ws, size_t ws_size,
                              hipStream_t stream)
{
  const float* x    = (const float*)d_in[0];
  const float* c1w  = (const float*)d_in[1];  const float* c1b = (const float*)d_in[2];
  const float* b1g  = (const float*)d_in[3];  const float* b1b = (const float*)d_in[4];
  const float* b1m  = (const float*)d_in[5];  const float* b1v = (const float*)d_in[6];
  const float* c2w  = (const float*)d_in[7];  const float* c2b = (const float*)d_in[8];
  const float* b2g  = (const float*)d_in[9];  const float* b2b = (const float*)d_in[10];
  const float* b2m  = (const float*)d_in[11]; const float* b2v = (const float*)d_in[12];
  const float* c3w  = (const float*)d_in[13]; const float* c3b = (const float*)d_in[14];
  const float* b3g  = (const float*)d_in[15]; const float* b3b = (const float*)d_in[16];
  const float* b3m  = (const float*)d_in[17]; const float* b3v = (const float*)d_in[18];
  const float* pf1w = (const float*)d_in[19]; const float* pf1b = (const float*)d_in[20];
  const float* plng = (const float*)d_in[21]; const float* plnb = (const float*)d_in[22];
  const float* pf2w = (const float*)d_in[23]; const float* pf2b = (const float*)d_in[24];
  const float* mf1w = (const float*)d_in[25]; const float* mf1b = (const float*)d_in[26];
  const float* mlng = (const float*)d_in[27]; const float* mlnb = (const float*)d_in[28];
  const float* mf2w = (const float*)d_in[29]; const float* mf2b = (const float*)d_in[30];
  const float* vf1w = (const float*)d_in[31]; const float* vf1b = (const float*)d_in[32];
  const float* vlng = (const float*)d_in[33]; const float* vlnb = (const float*)d_in[34];
  const float* vf2w = (const float*)d_in[35]; const float* vf2b = (const float*)d_in[36];

  float* ws = (float*)d_ws;
  // conv activations
  float* h1    = ws;              // 8192*64*36  = 18,874,368 f
  float* h2    = ws + 18874368;   // 8192*128*36 = 37,748,736 f
  float* feats = ws + 56623104;   // 8192*128*36 (== feats, row stride 4608)
  // FC scratch overlaid on h1 (dead after conv2)
  float* p1    = ws + 0;          // 8192*256
  float* plog  = ws + 2097152;    // 8192*36
  float* m1    = ws + 2392064;    // 8192*256
  float* mlog  = ws + 4489216;    // 8192*36
  float* v1    = ws + 4784128;    // 8192*256
  float* valb  = ws + 6881280;    // 8192
  int*   selb  = (int*)(ws + 6889472); // 8192
  float* out   = (float*)d_out;   // (8192, 1296) fp32

  const dim3 blk(256);
  const dim3 gConv(9216);         // 294912 rows / 32 per block

  conv_bn_relu_wmma<<<gConv, blk, 0, stream>>>(x,  c1w, c1b, b1g, b1b, b1m, b1v, h1,    4,  64);
  conv_bn_relu_wmma<<<gConv, blk, 0, stream>>>(h1, c2w, c2b, b2g, b2b, b2m, b2v, h2,   64, 128);
  conv_bn_relu_wmma<<<gConv, blk, 0, stream>>>(h2, c3w, c3b, b3g, b3b, b3m, b3v, feats, 128, 128);

  // piece head
  fc_wmma<<<dim3(64, 8), blk, 0, stream>>>(feats, pf1w, pf1b, nullptr, nullptr, p1, 256, 4608, 4608);
  ln_relu<<<dim3(1024), blk, 0, stream>>>(p1, plng, plnb);
  fc_wmma<<<dim3(64, 2), blk, 0, stream>>>(p1, pf2w, pf2b, nullptr, nullptr, plog, 36, 256, 256);
  argmax36<<<dim3(1024), blk, 0, stream>>>(plog, selb);

  // move head (one-hot concat folded into epilogue gather of mfc1_w column)
  fc_wmma<<<dim3(64, 8), blk, 0, stream>>>(feats, mf1w, mf1b, mf1w + 4608, selb, m1, 256, 4608, 4644);
  ln_relu<<<dim3(1024), blk, 0, stream>>>(m1, mlng, mlnb);
  fc_wmma<<<dim3(64, 2), blk, 0, stream>>>(m1, mf2w, mf2b, nullptr, nullptr, mlog, 36, 256, 256);

  // value head
  fc_wmma<<<dim3(64, 8), blk, 0, stream>>>(feats, vf1w, vf1b, nullptr, nullptr, v1, 256, 4608, 4608);
  ln_relu<<<dim3(1024), blk, 0, stream>>>(v1, vlng, vlnb);
  vfc2_dot<<<dim3(1024), blk, 0, stream>>>(v1, vf2w, vf2b, valb);

  dueling_combine<<<dim3(1024), blk, 0, stream>>>(plog, mlog, valb, out);

  (void)in_sizes; (void)n_in; (void)out_size; (void)ws_size;
}